// Model_69767448756496
// MI455X (gfx1250) — compile-verified
//
#include <hip/hip_runtime.h>
#include <stdint.h>

// out[i] = (mask[i] && pos[i] < n_upd) ? updates[pos[i]] : x[i]
// Pure streaming kernel: 128-bit NT loads/stores, branchless clamped gather.
// position is an exclusive prefix-sum of mask -> gather is monotone/coalesced.

typedef float v4f __attribute__((ext_vector_type(4)));
typedef int   v4i __attribute__((ext_vector_type(4)));

__global__ __launch_bounds__(256) void masked_prefix_gather_v4(
    const float* __restrict__ x,
    const unsigned char* __restrict__ mask, // numpy bool: 1 byte/elem
    const int* __restrict__ pos,            // JAX x64 off -> int32
    const float* __restrict__ upd,
    float* __restrict__ out,
    int n_upd,
    long long nquad)
{
    long long q = (long long)blockIdx.x * blockDim.x + threadIdx.x;
    if (q >= nquad) return;

    const v4f xv = __builtin_nontemporal_load((const v4f*)x + q);
    const v4i pv = __builtin_nontemporal_load((const v4i*)pos + q);
    const uint32_t mv = __builtin_nontemporal_load((const uint32_t*)mask + q);

    const int last = n_upd - 1;

    // Speculative prefetch ~4K elements ahead in the updates stream.
    // Positions are monotone, so this walks linearly ahead of the gather.
    {
        int pf = pv[3] + 4096;
        pf = pf > last ? last : pf;
        __builtin_prefetch(upd + pf, /*rw=*/0, /*locality=*/0); // global_prefetch_b8
    }

    int p0 = pv[0] > last ? last : pv[0];
    int p1 = pv[1] > last ? last : pv[1];
    int p2 = pv[2] > last ? last : pv[2];
    int p3 = pv[3] > last ? last : pv[3];

    // Unconditional gather: for invalid lanes the clamped index aliases a
    // neighbor's line already in flight, so this stays branch-free for ~free.
    float g0 = upd[p0];
    float g1 = upd[p1];
    float g2 = upd[p2];
    float g3 = upd[p3];

    v4f o;
    o[0] = ((mv & 0x000000ffu) && pv[0] < n_upd) ? g0 : xv[0];
    o[1] = ((mv & 0x0000ff00u) && pv[1] < n_upd) ? g1 : xv[1];
    o[2] = ((mv & 0x00ff0000u) && pv[2] < n_upd) ? g2 : xv[2];
    o[3] = ((mv & 0xff000000u) && pv[3] < n_upd) ? g3 : xv[3];

    __builtin_nontemporal_store(o, (v4f*)out + q);
}

// Scalar tail (only launched when n % 4 != 0; with 2^25 elements it is not).
__global__ __launch_bounds__(64) void masked_prefix_gather_tail(
    const float* __restrict__ x,
    const unsigned char* __restrict__ mask,
    const int* __restrict__ pos,
    const float* __restrict__ upd,
    float* __restrict__ out,
    int n_upd,
    long long start,
    long long n)
{
    long long i = start + (long long)blockIdx.x * blockDim.x + threadIdx.x;
    if (i >= n) return;
    int p = pos[i];
    int last = n_upd - 1;
    int ps = p > last ? last : p;
    float g = upd[ps];
    out[i] = (mask[i] && p < n_upd) ? g : x[i];
}

extern "C" void kernel_launch(void* const* d_in, const int* in_sizes, int n_in,
                              void* d_out, int out_size, void* d_ws, size_t ws_size,
                              hipStream_t stream) {
    const float*         x    = (const float*)d_in[0];
    const unsigned char* mask = (const unsigned char*)d_in[1];
    const int*           pos  = (const int*)d_in[2];
    const float*         upd  = (const float*)d_in[3];
    float*               out  = (float*)d_out;

    const long long n     = (long long)in_sizes[0];   // 32*1024*1024
    const int       n_upd = in_sizes[3];              // 16777216

    const long long nquad = n >> 2;
    if (nquad > 0) {
        const int block = 256;
        const long long grid = (nquad + block - 1) / block;
        masked_prefix_gather_v4<<<(unsigned)grid, block, 0, stream>>>(
            x, mask, pos, upd, out, n_upd, nquad);
    }

    const long long rem = n - (nquad << 2);
    if (rem > 0) {
        masked_prefix_gather_tail<<<(unsigned)((rem + 63) / 64), 64, 0, stream>>>(
            x, mask, pos, upd, out, n_upd, nquad << 2, n);
    }
}